// INTERVAL_82566451299183
// MI455X (gfx1250) — compile-verified
//
#include <hip/hip_runtime.h>
#include <hip/hip_bf16.h>
#include <math.h>

// Problem constants (match reference)
#define BB 4
#define LL 64
#define DD 64
#define RELN 16
#define NEGV (-1e9f)

typedef __attribute__((ext_vector_type(2))) float v2f;
typedef __attribute__((ext_vector_type(8))) float v8f;

// ---------------------------------------------------------------------------
// f32 WMMA 16x16x4 helper.  A: 16x4 (2 VGPRs/lane), B: 4x16 (2 VGPRs/lane),
// C/D: 16x16 (8 VGPRs/lane).  Layout per CDNA5 ISA 7.12.2:
//   A[m][k]: lane = (k>>1)*16 + m, vgpr = k&1
//   C[r][c]: lane = (r>=8)*16 + c, vgpr = r&7   (row = v + 8*(lane>>4))
// ---------------------------------------------------------------------------
__device__ inline v8f wmma_f32_4(v2f a, v2f b, v8f c) {
    return __builtin_amdgcn_wmma_f32_16x16x4_f32(
        /*neg_a=*/false, a, /*neg_b=*/false, b,
        /*c_mod=*/(short)0, c, /*reuse_a=*/false, /*reuse_b=*/false);
}

// ---------------------------------------------------------------------------
// Kernel 1: embedding gather   O[b,l,:] = emb_table[x[b,l],:]
// ---------------------------------------------------------------------------
__global__ void k_embed(const int* __restrict__ x, const float* __restrict__ emb,
                        float* __restrict__ O) {
    int bl = blockIdx.x;           // 0..255
    int d  = threadIdx.x;          // 0..63
    int idx = x[bl];
    O[bl * DD + d] = emb[idx * DD + d];
}

// ---------------------------------------------------------------------------
// Kernel 2: U[b,m] = O_b (64x64) @ M_m (64x64), all 16 relation matrices.
// Block = (b,m), 128 threads = 4 waves; each wave owns a 16-row stripe and
// sweeps 4 column tiles; K=64 -> 16 WMMA f32 16x16x4 issues per tile.
// Operands staged through LDS.
// ---------------------------------------------------------------------------
__global__ void k_relgemm(const float* __restrict__ O, const float* __restrict__ relT,
                          float* __restrict__ U) {
    __shared__ float sA[DD * DD];
    __shared__ float sB[DD * DD];
    int b = blockIdx.x >> 4;
    int m = blockIdx.x & 15;
    const float* A  = O    + b * DD * DD;
    const float* Bm = relT + m * DD * DD;
    for (int i = threadIdx.x; i < DD * DD; i += blockDim.x) { sA[i] = A[i]; sB[i] = Bm[i]; }
    __syncthreads();

    int wave = threadIdx.x >> 5;      // 0..3 -> row tile
    int lane = threadIdx.x & 31;
    int mr = lane & 15, half = lane >> 4;
    float* Ub = U + (size_t)((b * RELN + m) * DD) * DD;

    for (int tn = 0; tn < 4; ++tn) {
        v8f c = {};
        for (int kc = 0; kc < 16; ++kc) {
            int k0 = kc * 4 + 2 * half;
            v2f af, bf;
            af.x = sA[(wave * 16 + mr) * DD + k0];
            af.y = sA[(wave * 16 + mr) * DD + k0 + 1];
            bf.x = sB[(k0)     * DD + tn * 16 + mr];
            bf.y = sB[(k0 + 1) * DD + tn * 16 + mr];
            c = wmma_f32_4(af, bf, c);
        }
        for (int v = 0; v < 8; ++v) {
            int row = wave * 16 + v + 8 * half;
            Ub[row * DD + tn * 16 + mr] = c[v];
        }
    }
}

// ---------------------------------------------------------------------------
// Kernel 3: relation attention.  Block per (b,i), 64 threads.
//   scores[j] = r>0 ? dot(U[b,r,i,:], O[b,j,:]) : NEG ; softmax over j;
//   o' = p @ O_b ; ATT = o' @ attn_out_W^T + b
// ---------------------------------------------------------------------------
__global__ void k_attn(const float* __restrict__ O, const float* __restrict__ U,
                       const int* __restrict__ r, const float* __restrict__ W,
                       const float* __restrict__ bias, float* __restrict__ ATT) {
    __shared__ float sc[LL];
    __shared__ float red2[2];
    __shared__ float oo[DD];
    int b = blockIdx.x >> 6;
    int i = blockIdx.x & 63;
    int j = threadIdx.x;

    int rj = r[(b * LL + i) * LL + j];
    float s = NEGV;
    if (rj > 0) {
        const float* u  = U + (size_t)((b * RELN + rj) * LL + i) * DD;
        const float* ok = O + (b * LL + j) * DD;
        float acc = 0.f;
        for (int e = 0; e < DD; ++e) acc += u[e] * ok[e];
        s = acc;
    }
    sc[j] = s;
    __syncthreads();
    if (j == 0) {
        float mx = sc[0];
        for (int t = 1; t < LL; ++t) mx = fmaxf(mx, sc[t]);
        float sm = 0.f;
        for (int t = 0; t < LL; ++t) sm += expf(sc[t] - mx);
        red2[0] = mx; red2[1] = sm;
    }
    __syncthreads();
    float p = expf(s - red2[0]) / red2[1];
    sc[j] = p;
    __syncthreads();

    int d = threadIdx.x;
    float od = 0.f;
    for (int jj = 0; jj < LL; ++jj) od += sc[jj] * O[(b * LL + jj) * DD + d];
    oo[d] = od;
    __syncthreads();
    float out = bias[d];
    for (int e = 0; e < DD; ++e) out += oo[e] * W[d * DD + e];
    ATT[(b * LL + i) * DD + d] = out;
}

// ---------------------------------------------------------------------------
// Kernel 4: GX = ATT (256x64) @ Wih^T (64x192) + bih, via WMMA.
// 192 output tiles; 48 blocks x 4 waves, one tile per wave.
// ---------------------------------------------------------------------------
__global__ void k_gx(const float* __restrict__ ATT, const float* __restrict__ Wih,
                     const float* __restrict__ bih, float* __restrict__ GX) {
    int wave = threadIdx.x >> 5, lane = threadIdx.x & 31;
    int tile = blockIdx.x * 4 + wave;          // 0..191
    int tmi = tile / 12, tni = tile % 12;
    int mr = lane & 15, half = lane >> 4;
    v8f c = {};
    for (int kc = 0; kc < 16; ++kc) {
        int k0 = kc * 4 + 2 * half;
        v2f af, bf;
        int row = tmi * 16 + mr;
        af.x = ATT[row * DD + k0];
        af.y = ATT[row * DD + k0 + 1];
        int col = tni * 16 + mr;               // B[k][n] = Wih[n][k]
        bf.x = Wih[col * DD + k0];
        bf.y = Wih[col * DD + k0 + 1];
        c = wmma_f32_4(af, bf, c);
    }
    for (int v = 0; v < 8; ++v) {
        int row = tmi * 16 + v + 8 * half;
        int col = tni * 16 + mr;
        GX[row * 192 + col] = c[v] + bih[col];
    }
}

// ---------------------------------------------------------------------------
// Kernel 5: GRU recurrence.  Single workgroup, 384 threads = 12 waves.
// h held zero-padded to 16 rows in LDS; each wave computes one 16-col tile of
// gh = h @ Whh^T per timestep via 16 WMMA issues; gate math on 256 threads.
// ---------------------------------------------------------------------------
__global__ void k_gru(const float* __restrict__ GX, const float* __restrict__ Whh,
                      const float* __restrict__ bhh, float* __restrict__ GO) {
    __shared__ float hP[16 * DD];      // padded hidden state (rows 4..15 zero)
    __shared__ float gh[BB * 192];
    for (int i = threadIdx.x; i < 16 * DD; i += blockDim.x) hP[i] = 0.f;
    __syncthreads();

    int wave = threadIdx.x >> 5, lane = threadIdx.x & 31;
    int mr = lane & 15, half = lane >> 4;
    int tn = wave;                      // 0..11 column tile of 192

    for (int t = 0; t < LL; ++t) {
        v8f c = {};
        for (int kc = 0; kc < 16; ++kc) {
            int k0 = kc * 4 + 2 * half;
            v2f af, bf;
            af.x = hP[mr * DD + k0];
            af.y = hP[mr * DD + k0 + 1];
            int col = tn * 16 + mr;     // B[k][n] = Whh[n][k]
            bf.x = Whh[col * DD + k0];
            bf.y = Whh[col * DD + k0 + 1];
            c = wmma_f32_4(af, bf, c);
        }
        for (int v = 0; v < 8; ++v) {
            int row = v + 8 * half;
            if (row < BB) gh[row * 192 + tn * 16 + mr] = c[v];
        }
        __syncthreads();
        if (threadIdx.x < BB * DD) {
            int b = threadIdx.x >> 6, d = threadIdx.x & 63;
            const float* g = GX + (b * LL + t) * 192;
            float xr = g[d], xz = g[64 + d], xn = g[128 + d];
            float hr = gh[b * 192 + d]       + bhh[d];
            float hz = gh[b * 192 + 64 + d]  + bhh[64 + d];
            float hn = gh[b * 192 + 128 + d] + bhh[128 + d];
            float rg = 1.f / (1.f + expf(-(xr + hr)));
            float zg = 1.f / (1.f + expf(-(xz + hz)));
            float ng = tanhf(xn + rg * hn);
            float h  = hP[b * DD + d];
            float hn2 = (1.f - zg) * ng + zg * h;
            hP[b * DD + d] = hn2;
            GO[(b * LL + t) * DD + d] = hn2;
        }
        __syncthreads();
    }
}

// ---------------------------------------------------------------------------
// Kernel 6: output GRU (H=1) -> masked softmax pooling -> classifier -> loss.
// Single workgroup, 256 threads.  d_out = [y_hat (4x2), loss (1)].
// ---------------------------------------------------------------------------
__global__ void k_final(const float* __restrict__ GO, const int* __restrict__ l,
                        const int* __restrict__ y,
                        const float* __restrict__ oWih, const float* __restrict__ oWhh,
                        const float* __restrict__ obih, const float* __restrict__ obhh,
                        const float* __restrict__ W1, const float* __restrict__ b1,
                        const float* __restrict__ W2, const float* __restrict__ b2,
                        float* __restrict__ out) {
    __shared__ float g3[BB * LL * 3];
    __shared__ float ga[BB * LL];
    __shared__ float redm[BB], reds[BB];
    __shared__ float cbd[BB * DD];
    __shared__ float h1[BB * 32];
    __shared__ float lg[BB * 2];
    __shared__ float pl[BB];
    __shared__ int lmax_s;

    if (threadIdx.x == 0) {
        int lm = l[0];
        for (int i = 1; i < BB; ++i) lm = max(lm, l[i]);
        lmax_s = lm;                       // mask cols < max(l)+1 (torch loop bug)
    }
    // gx2[b,t,g] = GO[b,t,:] . oWih[g,:] + obih[g]
    for (int idx = threadIdx.x; idx < BB * LL * 3; idx += blockDim.x) {
        int g = idx % 3, bt = idx / 3;
        const float* row = GO + bt * DD;
        float acc = obih[g];
        for (int e = 0; e < DD; ++e) acc += row[e] * oWih[g * DD + e];
        g3[bt * 3 + g] = acc;
    }
    __syncthreads();
    if (threadIdx.x < BB) {                // sequential H=1 GRU per batch
        int b = threadIdx.x;
        float h = 0.f;
        for (int t = 0; t < LL; ++t) {
            const float* g = &g3[(b * LL + t) * 3];
            float hr = h * oWhh[0] + obhh[0];
            float hz = h * oWhh[1] + obhh[1];
            float hn = h * oWhh[2] + obhh[2];
            float rg = 1.f / (1.f + expf(-(g[0] + hr)));
            float zg = 1.f / (1.f + expf(-(g[1] + hz)));
            float ng = tanhf(g[2] + rg * hn);
            h = (1.f - zg) * ng + zg * h;
            ga[b * LL + t] = h;
        }
    }
    __syncthreads();
    {   // mask
        int b = threadIdx.x >> 6, t = threadIdx.x & 63;
        float v = (t < lmax_s + 1) ? ga[b * LL + t] : NEGV;
        ga[b * LL + t] = v;
    }
    __syncthreads();
    if (threadIdx.x < BB) {
        int b = threadIdx.x;
        float mx = ga[b * LL];
        for (int t = 1; t < LL; ++t) mx = fmaxf(mx, ga[b * LL + t]);
        float sm = 0.f;
        for (int t = 0; t < LL; ++t) sm += expf(ga[b * LL + t] - mx);
        redm[b] = mx; reds[b] = sm;
    }
    __syncthreads();
    {   // normalize
        int b = threadIdx.x >> 6, t = threadIdx.x & 63;
        ga[b * LL + t] = expf(ga[b * LL + t] - redm[b]) / reds[b];
    }
    __syncthreads();
    {   // pooled context c[b,d]
        int b = threadIdx.x >> 6, d = threadIdx.x & 63;
        float acc = 0.f;
        for (int t = 0; t < LL; ++t) acc += ga[b * LL + t] * GO[(b * LL + t) * DD + d];
        cbd[b * DD + d] = acc;
    }
    __syncthreads();
    if (threadIdx.x < BB * 32) {           // hidden layer, relu
        int b = threadIdx.x >> 5, k = threadIdx.x & 31;
        float acc = b1[k];
        for (int e = 0; e < DD; ++e) acc += cbd[b * DD + e] * W1[k * DD + e];
        h1[b * 32 + k] = fmaxf(acc, 0.f);
    }
    __syncthreads();
    if (threadIdx.x < BB * 2) {            // logits
        int b = threadIdx.x >> 1, cc = threadIdx.x & 1;
        float acc = b2[cc];
        for (int k = 0; k < 32; ++k) acc += h1[b * 32 + k] * W2[cc * 32 + k];
        lg[b * 2 + cc] = acc;
    }
    __syncthreads();
    if (threadIdx.x < BB) {                // softmax + per-sample logp
        int b = threadIdx.x;
        float l0 = lg[b * 2], l1 = lg[b * 2 + 1];
        float m = fmaxf(l0, l1);
        float e0 = expf(l0 - m), e1 = expf(l1 - m), s = e0 + e1;
        out[b * 2]     = e0 / s;
        out[b * 2 + 1] = e1 / s;
        int yb = y[b];
        pl[b] = (yb ? l1 : l0) - m - logf(s);
    }
    __syncthreads();
    if (threadIdx.x == 0) out[BB * 2] = -(pl[0] + pl[1] + pl[2] + pl[3]) * 0.25f;
}

// ---------------------------------------------------------------------------
extern "C" void kernel_launch(void* const* d_in, const int* in_sizes, int n_in,
                              void* d_out, int out_size, void* d_ws, size_t ws_size,
                              hipStream_t stream) {
    const int*   x     = (const int*)  d_in[0];
    const int*   y     = (const int*)  d_in[1];
    const int*   r     = (const int*)  d_in[2];
    const int*   l     = (const int*)  d_in[3];
    const float* emb   = (const float*)d_in[4];
    const float* relT  = (const float*)d_in[5];
    const float* attnW = (const float*)d_in[6];
    const float* attnB = (const float*)d_in[7];
    const float* gWih  = (const float*)d_in[8];
    const float* gWhh  = (const float*)d_in[9];
    const float* gbih  = (const float*)d_in[10];
    const float* gbhh  = (const float*)d_in[11];
    const float* oWih  = (const float*)d_in[12];
    const float* oWhh  = (const float*)d_in[13];
    const float* obih  = (const float*)d_in[14];
    const float* obhh  = (const float*)d_in[15];
    const float* W1    = (const float*)d_in[16];
    const float* b1    = (const float*)d_in[17];
    const float* W2    = (const float*)d_in[18];
    const float* b2    = (const float*)d_in[19];

    float* ws  = (float*)d_ws;
    float* O   = ws;                        // 16384
    float* U   = O   + BB * LL * DD;        // 262144
    float* ATT = U   + BB * RELN * LL * DD; // 16384
    float* GX  = ATT + BB * LL * DD;        // 49152
    float* GO  = GX  + BB * LL * 192;       // 16384

    k_embed  <<<BB * LL, DD,      0, stream>>>(x, emb, O);
    k_relgemm<<<BB * RELN, 128,   0, stream>>>(O, relT, U);
    k_attn   <<<BB * LL, LL,      0, stream>>>(O, U, r, attnW, attnB, ATT);
    k_gx     <<<48, 128,          0, stream>>>(ATT, gWih, gbih, GX);
    k_gru    <<<1, 384,           0, stream>>>(GX, gWhh, gbhh, GO);
    k_final  <<<1, 256,           0, stream>>>(GO, l, y, oWih, oWhh, obih, obhh,
                                               W1, b1, W2, b2, (float*)d_out);
}